// OptimizedGNN_9302899163675
// MI455X (gfx1250) — compile-verified
//
#include <hip/hip_runtime.h>
#include <math.h>

#define N_NODES 50000
#define N_EDGES 800000
#define N_GRAPHS 512
#define F_IN 128
#define HID 64
#define H1C 256   // 4 heads * 64 = layer-1 channels

typedef __attribute__((ext_vector_type(2))) float v2f;
typedef __attribute__((ext_vector_type(8))) float v8f;

// ---------------- utility device functions ----------------

__device__ __forceinline__ float warp_sum32(float v) {
#pragma unroll
  for (int m = 1; m < 32; m <<= 1) v += __shfl_xor(v, m, 32);
  return v;
}

__device__ __forceinline__ float group8_sum(float v) {
  v += __shfl_xor(v, 1, 32);
  v += __shfl_xor(v, 2, 32);
  v += __shfl_xor(v, 4, 32);
  return v;
}

// float atomic max via sign-aware int/uint atomics (lowers to
// global_atomic_max_i32 / global_atomic_min_u32 — native CDNA5 ops).
// Buffer must be initialized to -INF.
__device__ __forceinline__ void atomicMaxF(float* addr, float val) {
  if (val >= 0.0f) {
    atomicMax((int*)addr, __float_as_int(val));
  } else {
    atomicMin((unsigned int*)addr, __float_as_uint(val));
  }
}

// ---------------- fill kernel ----------------

__global__ void fill_kernel(float* __restrict__ p, float v, int n) {
  int i = blockIdx.x * blockDim.x + threadIdx.x;
  if (i < n) p[i] = v;
}

// ---------------- WMMA f32 GEMM:  C[M,Nw] = A[M,K] @ W[K,Nw] + bias ----------------
// One wave computes a 16x64 strip (4 accumulators) via v_wmma_f32_16x16x4_f32,
// reusing the A fragment across the 4 N-subtiles (2 A loads + 8 B loads -> 4 WMMA
// per K-step). M multiple of 16, Nw multiple of 64 (or handled by wave guard),
// K multiple of 4.
// A fragment (16x4): VGPR0 = {K+0 lanes 0-15, K+2 lanes 16-31}, VGPR1 = {K+1, K+3}
// B fragment (4x16): same K split, N = lane&15
// C/D (16x16): VGPR r = {M=r lanes 0-15, M=r+8 lanes 16-31}, N = lane&15
__global__ void wmma_gemm_bias(const float* __restrict__ A, int K,
                               const float* __restrict__ W, int Nw,
                               const float* __restrict__ bias,
                               float* __restrict__ C) {
  const int lane = threadIdx.x & 31;
  const int wave = threadIdx.x >> 5;
  const int m0 = blockIdx.x * 16;
  const int n0 = (blockIdx.y * 4 + wave) * 64;   // 64-wide strip per wave
  if (n0 >= Nw) return;  // wave-uniform guard; EXEC stays all-ones
  const int hi = lane >> 4;   // 0: lanes 0-15, 1: lanes 16-31
  const int l = lane & 15;

  v8f c0 = {0.f, 0.f, 0.f, 0.f, 0.f, 0.f, 0.f, 0.f};
  v8f c1 = c0, c2 = c0, c3 = c0;
  const float* arow = A + (size_t)(m0 + l) * K;

  for (int k = 0; k < K; k += 4) {
    v2f a;
    a[0] = arow[k + hi * 2];
    a[1] = arow[k + hi * 2 + 1];
    const float* w0 = W + (size_t)(k + hi * 2) * Nw + n0 + l;
    const float* w1 = W + (size_t)(k + hi * 2 + 1) * Nw + n0 + l;
    v2f b0, b1, b2, b3;
    b0[0] = w0[0];  b0[1] = w1[0];
    b1[0] = w0[16]; b1[1] = w1[16];
    b2[0] = w0[32]; b2[1] = w1[32];
    b3[0] = w0[48]; b3[1] = w1[48];
    c0 = __builtin_amdgcn_wmma_f32_16x16x4_f32(false, a, false, b0, (short)0, c0, false, false);
    c1 = __builtin_amdgcn_wmma_f32_16x16x4_f32(false, a, false, b1, (short)0, c1, false, false);
    c2 = __builtin_amdgcn_wmma_f32_16x16x4_f32(false, a, false, b2, (short)0, c2, false, false);
    c3 = __builtin_amdgcn_wmma_f32_16x16x4_f32(false, a, false, b3, (short)0, c3, false, false);
  }

  const float bv0 = bias ? bias[n0 + l]      : 0.0f;
  const float bv1 = bias ? bias[n0 + 16 + l] : 0.0f;
  const float bv2 = bias ? bias[n0 + 32 + l] : 0.0f;
  const float bv3 = bias ? bias[n0 + 48 + l] : 0.0f;
#pragma unroll
  for (int r = 0; r < 8; ++r) {
    const int row = m0 + r + hi * 8;
    float* crow = C + (size_t)row * Nw + n0 + l;
    crow[0]  = c0[r] + bv0;
    crow[16] = c1[r] + bv1;
    crow[32] = c2[r] + bv2;
    crow[48] = c3[r] + bv3;
  }
}

// ---------------- layer 1 edge kernels (heads=4, dim=64, ch=256) ----------------

// wave per edge; lane covers 8 channels; per-head (8-lane group) dot product
__global__ void edge_logits1(const int* __restrict__ ei, const float* __restrict__ eattr,
                             const float* __restrict__ q, const float* __restrict__ k,
                             const float* __restrict__ We,
                             float* __restrict__ pbuf, float* __restrict__ mbuf) {
  const int lane = threadIdx.x & 31;
  const int e = blockIdx.x * 8 + (threadIdx.x >> 5);
  if (e >= N_EDGES) return;
  const int s = ei[e];
  const int t = ei[N_EDGES + e];
  const float ea = eattr[e];
  const float* qr = q + (size_t)t * H1C;
  const float* kr = k + (size_t)s * H1C;
  const int c0 = lane * 8;
  float part = 0.f;
#pragma unroll
  for (int j = 0; j < 8; ++j) {
    const int c = c0 + j;
    part += qr[c] * (kr[c] + ea * We[c]);
  }
  part = group8_sum(part);
  if ((lane & 7) == 0) {
    const int h = lane >> 3;
    const float logit = part * 0.125f;  // / sqrt(64)
    pbuf[(size_t)e * 4 + h] = logit;
    atomicMaxF(&mbuf[(size_t)t * 4 + h], logit);
  }
}

__global__ void edge_expsum1(const int* __restrict__ ei, float* __restrict__ pbuf,
                             const float* __restrict__ mbuf, float* __restrict__ sbuf) {
  const int i = blockIdx.x * blockDim.x + threadIdx.x;
  if (i >= N_EDGES * 4) return;
  const int e = i >> 2, h = i & 3;
  const int t = ei[N_EDGES + e];
  const float p = expf(pbuf[i] - mbuf[(size_t)t * 4 + h]);
  pbuf[i] = p;
  atomicAdd(&sbuf[(size_t)t * 4 + h], p);
}

__global__ void edge_scatter1(const int* __restrict__ ei, const float* __restrict__ eattr,
                              const float* __restrict__ v, const float* __restrict__ We,
                              const float* __restrict__ pbuf, float* __restrict__ acc) {
  const int lane = threadIdx.x & 31;
  const int e = blockIdx.x * 8 + (threadIdx.x >> 5);
  if (e >= N_EDGES) return;
  const int s = ei[e];
  const int t = ei[N_EDGES + e];
  const float ea = eattr[e];
  const float p = pbuf[(size_t)e * 4 + (lane >> 3)];
  const float* vr = v + (size_t)s * H1C;
  float* ar = acc + (size_t)t * H1C;
  const int c0 = lane * 8;
#pragma unroll
  for (int j = 0; j < 8; ++j) {
    const int c = c0 + j;
    atomicAdd(&ar[c], p * (vr[c] + ea * We[c]));
  }
}

// wave per node: out = LN(elu(acc/sum + skip)) over 256 channels (8/lane)
__global__ void finalize1(const float* __restrict__ acc, const float* __restrict__ sbuf,
                          const float* __restrict__ skip,
                          const float* __restrict__ g, const float* __restrict__ b,
                          float* __restrict__ out) {
  const int lane = threadIdx.x & 31;
  const int n = blockIdx.x * 8 + (threadIdx.x >> 5);
  if (n >= N_NODES) return;
  const float sden = sbuf[(size_t)n * 4 + (lane >> 3)] + 1e-16f;
  const int c0 = lane * 8;
  float vals[8];
  float psum = 0.f;
#pragma unroll
  for (int j = 0; j < 8; ++j) {
    const int c = c0 + j;
    float x = acc[(size_t)n * H1C + c] / sden + skip[(size_t)n * H1C + c];
    x = x > 0.f ? x : (expf(x) - 1.f);  // ELU
    vals[j] = x;
    psum += x;
  }
  const float mean = warp_sum32(psum) * (1.0f / H1C);
  float vsum = 0.f;
#pragma unroll
  for (int j = 0; j < 8; ++j) {
    const float d = vals[j] - mean;
    vsum += d * d;
  }
  const float rstd = rsqrtf(warp_sum32(vsum) * (1.0f / H1C) + 1e-5f);
#pragma unroll
  for (int j = 0; j < 8; ++j) {
    const int c = c0 + j;
    out[(size_t)n * H1C + c] = (vals[j] - mean) * rstd * g[c] + b[c];
  }
}

// ---------------- layer 2 edge kernels (heads=1, dim=64) ----------------

__global__ void edge_logits2(const int* __restrict__ ei, const float* __restrict__ eattr,
                             const float* __restrict__ q, const float* __restrict__ k,
                             const float* __restrict__ We,
                             float* __restrict__ pbuf, float* __restrict__ mbuf) {
  const int lane = threadIdx.x & 31;
  const int e = blockIdx.x * 8 + (threadIdx.x >> 5);
  if (e >= N_EDGES) return;
  const int s = ei[e];
  const int t = ei[N_EDGES + e];
  const float ea = eattr[e];
  const float* qr = q + (size_t)t * HID;
  const float* kr = k + (size_t)s * HID;
  const int c = lane * 2;
  float part = qr[c] * (kr[c] + ea * We[c]) + qr[c + 1] * (kr[c + 1] + ea * We[c + 1]);
  part = warp_sum32(part);
  if (lane == 0) {
    const float logit = part * 0.125f;
    pbuf[e] = logit;
    atomicMaxF(&mbuf[t], logit);
  }
}

__global__ void edge_expsum2(const int* __restrict__ ei, float* __restrict__ pbuf,
                             const float* __restrict__ mbuf, float* __restrict__ sbuf) {
  const int e = blockIdx.x * blockDim.x + threadIdx.x;
  if (e >= N_EDGES) return;
  const int t = ei[N_EDGES + e];
  const float p = expf(pbuf[e] - mbuf[t]);
  pbuf[e] = p;
  atomicAdd(&sbuf[t], p);
}

__global__ void edge_scatter2(const int* __restrict__ ei, const float* __restrict__ eattr,
                              const float* __restrict__ v, const float* __restrict__ We,
                              const float* __restrict__ pbuf, float* __restrict__ acc) {
  const int lane = threadIdx.x & 31;
  const int e = blockIdx.x * 8 + (threadIdx.x >> 5);
  if (e >= N_EDGES) return;
  const int s = ei[e];
  const int t = ei[N_EDGES + e];
  const float ea = eattr[e];
  const float p = pbuf[e];
  const float* vr = v + (size_t)s * HID;
  float* ar = acc + (size_t)t * HID;
  const int c = lane * 2;
  atomicAdd(&ar[c], p * (vr[c] + ea * We[c]));
  atomicAdd(&ar[c + 1], p * (vr[c + 1] + ea * We[c + 1]));
}

// wave per node: h2 = LN(elu(acc/sum + skip)); fused mean/max pooling
__global__ void finalize2_pool(const float* __restrict__ acc, const float* __restrict__ sbuf,
                               const float* __restrict__ skip,
                               const float* __restrict__ g, const float* __restrict__ b,
                               const int* __restrict__ batch,
                               float* __restrict__ h2,
                               float* __restrict__ poolS, float* __restrict__ poolM,
                               float* __restrict__ cnt) {
  const int lane = threadIdx.x & 31;
  const int n = blockIdx.x * 8 + (threadIdx.x >> 5);
  if (n >= N_NODES) return;
  const float sden = sbuf[n] + 1e-16f;
  const int c = lane * 2;
  float x0 = acc[(size_t)n * HID + c] / sden + skip[(size_t)n * HID + c];
  float x1 = acc[(size_t)n * HID + c + 1] / sden + skip[(size_t)n * HID + c + 1];
  x0 = x0 > 0.f ? x0 : (expf(x0) - 1.f);
  x1 = x1 > 0.f ? x1 : (expf(x1) - 1.f);
  const float mean = warp_sum32(x0 + x1) * (1.0f / HID);
  const float d0 = x0 - mean, d1 = x1 - mean;
  const float rstd = rsqrtf(warp_sum32(d0 * d0 + d1 * d1) * (1.0f / HID) + 1e-5f);
  const float y0 = d0 * rstd * g[c] + b[c];
  const float y1 = d1 * rstd * g[c + 1] + b[c + 1];
  h2[(size_t)n * HID + c] = y0;
  h2[(size_t)n * HID + c + 1] = y1;
  const int bg = batch[n];
  atomicAdd(&poolS[(size_t)bg * HID + c], y0);
  atomicAdd(&poolS[(size_t)bg * HID + c + 1], y1);
  atomicMaxF(&poolM[(size_t)bg * HID + c], y0);
  atomicMaxF(&poolM[(size_t)bg * HID + c + 1], y1);
  if (lane == 0) atomicAdd(&cnt[bg], 1.0f);
}

// ---------------- final MLP: [G,128] -> relu(64) -> 1 ----------------

__global__ void mlp_kernel(const float* __restrict__ poolS, const float* __restrict__ poolM,
                           const float* __restrict__ cnt,
                           const float* __restrict__ w1, const float* __restrict__ b1,
                           const float* __restrict__ w2, const float* __restrict__ b2,
                           float* __restrict__ out) {
  __shared__ float sh[64];
  const int gidx = blockIdx.x;
  const int j = threadIdx.x;
  const float inv = 1.0f / fmaxf(cnt[gidx], 1.0f);
  float accv = b1[j];
  for (int i = 0; i < HID; ++i)
    accv += (poolS[(size_t)gidx * HID + i] * inv) * w1[i * HID + j];
  for (int i = 0; i < HID; ++i)
    accv += poolM[(size_t)gidx * HID + i] * w1[(HID + i) * HID + j];
  accv = fmaxf(accv, 0.0f);  // ReLU (dropout is identity in eval)
  sh[j] = accv * w2[j];
  __syncthreads();
  for (int st = 32; st > 0; st >>= 1) {
    if (j < st) sh[j] += sh[j + st];
    __syncthreads();
  }
  if (j == 0) out[gidx] = sh[0] + b2[0];
}

// ---------------- host launcher ----------------

extern "C" void kernel_launch(void* const* d_in, const int* in_sizes, int n_in,
                              void* d_out, int out_size, void* d_ws, size_t ws_size,
                              hipStream_t stream) {
  const float* x       = (const float*)d_in[0];
  const int*   ei      = (const int*)d_in[1];
  const float* eattr   = (const float*)d_in[2];
  const int*   batch   = (const int*)d_in[3];
  const float* q1_w = (const float*)d_in[4];  const float* q1_b = (const float*)d_in[5];
  const float* k1_w = (const float*)d_in[6];  const float* k1_b = (const float*)d_in[7];
  const float* v1_w = (const float*)d_in[8];  const float* v1_b = (const float*)d_in[9];
  const float* e1_w = (const float*)d_in[10];
  const float* s1_w = (const float*)d_in[11]; const float* s1_b = (const float*)d_in[12];
  const float* ln1_g = (const float*)d_in[13]; const float* ln1_b = (const float*)d_in[14];
  const float* q2_w = (const float*)d_in[15]; const float* q2_b = (const float*)d_in[16];
  const float* k2_w = (const float*)d_in[17]; const float* k2_b = (const float*)d_in[18];
  const float* v2_w = (const float*)d_in[19]; const float* v2_b = (const float*)d_in[20];
  const float* e2_w = (const float*)d_in[21];
  const float* s2_w = (const float*)d_in[22]; const float* s2_b = (const float*)d_in[23];
  const float* ln2_g = (const float*)d_in[24]; const float* ln2_b = (const float*)d_in[25];
  const float* m1_w = (const float*)d_in[26]; const float* m1_b = (const float*)d_in[27];
  const float* m2_w = (const float*)d_in[28]; const float* m2_b = (const float*)d_in[29];
  float* out = (float*)d_out;

  // ---- workspace layout (floats); layer-2 reuses layer-1 region ----
  float* base = (float*)d_ws;
  const size_t NB = (size_t)N_NODES * H1C;  // 12.8M floats
  float* q1   = base;
  float* k1   = base + NB;
  float* v1   = base + 2 * NB;
  float* sk1  = base + 3 * NB;
  float* acc1 = base + 4 * NB;
  float* h1   = base + 5 * NB;
  float* p1   = base + 6 * NB;                       // E*4
  float* mx1  = p1 + (size_t)N_EDGES * 4;            // N*4
  float* sm1  = mx1 + (size_t)N_NODES * 4;           // N*4
  // layer-2 region: reuse [base, base+5*NB); h1 stays live at 5*NB
  const size_t NH = (size_t)N_NODES * HID;           // 3.2M floats
  float* q2   = base;
  float* k2   = base + NH;
  float* v2   = base + 2 * NH;
  float* sk2  = base + 3 * NH;
  float* acc2 = base + 4 * NH;
  float* h2b  = base + 5 * NH;
  float* p2   = base + 6 * NH;                       // E
  float* mx2  = p2 + (size_t)N_EDGES;                // N
  float* sm2  = mx2 + (size_t)N_NODES;               // N
  float* poolS = sm2 + (size_t)N_NODES;              // G*64
  float* poolM = poolS + (size_t)N_GRAPHS * HID;     // G*64
  float* cnt   = poolM + (size_t)N_GRAPHS * HID;     // G

  const float NEG_INF = -__builtin_huge_valf();
  auto fill = [&](float* p, float v, size_t n) {
    fill_kernel<<<dim3((unsigned)((n + 255) / 256)), dim3(256), 0, stream>>>(p, v, (int)n);
  };

  // ===== layer 1 projections (WMMA GEMMs): [N,128] @ [128,256] + b =====
  // 4 waves/block, each owning a 16x64 strip -> whole 256-wide row per block
  const dim3 gemm1_grid(N_NODES / 16, 1);
  wmma_gemm_bias<<<gemm1_grid, 128, 0, stream>>>(x, F_IN, q1_w, H1C, q1_b, q1);
  wmma_gemm_bias<<<gemm1_grid, 128, 0, stream>>>(x, F_IN, k1_w, H1C, k1_b, k1);
  wmma_gemm_bias<<<gemm1_grid, 128, 0, stream>>>(x, F_IN, v1_w, H1C, v1_b, v1);
  wmma_gemm_bias<<<gemm1_grid, 128, 0, stream>>>(x, F_IN, s1_w, H1C, s1_b, sk1);

  fill(acc1, 0.0f, NB);
  fill(mx1, NEG_INF, (size_t)N_NODES * 4);
  fill(sm1, 0.0f, (size_t)N_NODES * 4);

  // ===== layer 1 attention =====
  const dim3 egrid((N_EDGES + 7) / 8);
  edge_logits1<<<egrid, 256, 0, stream>>>(ei, eattr, q1, k1, e1_w, p1, mx1);
  edge_expsum1<<<dim3((N_EDGES * 4 + 255) / 256), 256, 0, stream>>>(ei, p1, mx1, sm1);
  edge_scatter1<<<egrid, 256, 0, stream>>>(ei, eattr, v1, e1_w, p1, acc1);
  finalize1<<<dim3((N_NODES + 7) / 8), 256, 0, stream>>>(acc1, sm1, sk1, ln1_g, ln1_b, h1);

  // ===== layer 2 projections: [N,256] @ [256,64] + b =====
  // single-wave blocks: one 16x64 strip covers the full output width
  const dim3 gemm2_grid(N_NODES / 16, 1);
  wmma_gemm_bias<<<gemm2_grid, 32, 0, stream>>>(h1, H1C, q2_w, HID, q2_b, q2);
  wmma_gemm_bias<<<gemm2_grid, 32, 0, stream>>>(h1, H1C, k2_w, HID, k2_b, k2);
  wmma_gemm_bias<<<gemm2_grid, 32, 0, stream>>>(h1, H1C, v2_w, HID, v2_b, v2);
  wmma_gemm_bias<<<gemm2_grid, 32, 0, stream>>>(h1, H1C, s2_w, HID, s2_b, sk2);

  fill(acc2, 0.0f, NH);
  fill(mx2, NEG_INF, (size_t)N_NODES);
  fill(sm2, 0.0f, (size_t)N_NODES);
  fill(poolS, 0.0f, (size_t)N_GRAPHS * HID);
  fill(poolM, NEG_INF, (size_t)N_GRAPHS * HID);
  fill(cnt, 0.0f, (size_t)N_GRAPHS);

  // ===== layer 2 attention + pooling =====
  edge_logits2<<<egrid, 256, 0, stream>>>(ei, eattr, q2, k2, e2_w, p2, mx2);
  edge_expsum2<<<dim3((N_EDGES + 255) / 256), 256, 0, stream>>>(ei, p2, mx2, sm2);
  edge_scatter2<<<egrid, 256, 0, stream>>>(ei, eattr, v2, e2_w, p2, acc2);
  finalize2_pool<<<dim3((N_NODES + 7) / 8), 256, 0, stream>>>(acc2, sm2, sk2, ln2_g, ln2_b,
                                                              batch, h2b, poolS, poolM, cnt);

  // ===== MLP head =====
  mlp_kernel<<<dim3(N_GRAPHS), 64, 0, stream>>>(poolS, poolM, cnt, m1_w, m1_b, m2_w, m2_b, out);
}